// LaplaceKernel_28149215658537
// MI455X (gfx1250) — compile-verified
//
#include <hip/hip_runtime.h>
#include <hip/hip_bf16.h>

// Laplace (L1 / exponential) kernel matrix on gfx1250 (MI455X).
//
//   out[i,j] = exp(-max( (1/|ls|) * sum_d |x1[i,d] - x2[j,d]| , 1e-15 ))
//
// The reference's mean-adjustment cancels exactly inside the pairwise
// difference, so it is skipped (pure rounding-level difference, ~6e-6 rel).
//
// L1 distance is not expressible as A×B+C, so WMMA does not apply; the
// optimal formulation is 2 VALU ops per (i,j,d) term (v_sub_f32 +
// v_add_f32 with |src| modifier), register-blocked 8x8 per lane so the
// VALU:LDS ratio is 128:4 and gfx1250 dual-issue has abundant ILP.
// Output (256 MB, write-once) is streamed with non-temporal store hints
// so the L2-resident 4 MB of inputs are not evicted.

typedef float v4f __attribute__((ext_vector_type(4)));

#define TILE 128
#define DIMS 64

__global__ __launch_bounds__(256) void laplace_l1_exp_kernel(
    const float* __restrict__ x1,   // [N, 64]
    const float* __restrict__ x2,   // [M, 64]
    const float* __restrict__ ls,   // [1]
    float* __restrict__ out,        // [N, M]
    int N, int M)
{
    // Transposed tiles: feature-major so the inner loop reads contiguous
    // 32B chunks per lane (ds_load_b128, conflict-free).
    __shared__ __align__(16) float As[DIMS][TILE];  // 32 KB  As[d][i]
    __shared__ __align__(16) float Bs[DIMS][TILE];  // 32 KB  Bs[d][j]

    const int t  = threadIdx.x;
    const int i0 = blockIdx.y * TILE;
    const int j0 = blockIdx.x * TILE;

    // ---- Tile fill: coalesced float4 global loads, transposed LDS stores.
    // 256 threads, each moves half of one 64-float row (32 floats) of each tile.
    {
        const int r  = t >> 1;            // row within tile: 0..127
        const int c0 = (t & 1) * 32;      // feature offset: 0 or 32
        const v4f* g1 = (const v4f*)(x1 + (size_t)(i0 + r) * DIMS + c0);
        const v4f* g2 = (const v4f*)(x2 + (size_t)(j0 + r) * DIMS + c0);
        #pragma unroll
        for (int k = 0; k < 8; ++k) {
            v4f va = g1[k];
            v4f vb = g2[k];
            const int c = c0 + 4 * k;
            As[c + 0][r] = va.x; As[c + 1][r] = va.y;
            As[c + 2][r] = va.z; As[c + 3][r] = va.w;
            Bs[c + 0][r] = vb.x; Bs[c + 1][r] = vb.y;
            Bs[c + 2][r] = vb.z; Bs[c + 3][r] = vb.w;
        }
    }
    __syncthreads();

    // ---- 8x8 register micro-tile per lane.
    // tj from low 4 bits => lanes 0..15 of a wave cover one contiguous
    // 512B output row segment => coalesced NT stores.
    const int tj = (t & 15) * 8;   // j offset within tile
    const int ti = (t >> 4) * 8;   // i offset within tile

    float acc[8][8];
    #pragma unroll
    for (int a = 0; a < 8; ++a)
        #pragma unroll
        for (int b = 0; b < 8; ++b) acc[a][b] = 0.0f;

    #pragma unroll 4
    for (int d = 0; d < DIMS; ++d) {
        v4f a0 = *(const v4f*)&As[d][ti];
        v4f a1 = *(const v4f*)&As[d][ti + 4];
        v4f b0 = *(const v4f*)&Bs[d][tj];
        v4f b1 = *(const v4f*)&Bs[d][tj + 4];
        float av[8] = {a0.x, a0.y, a0.z, a0.w, a1.x, a1.y, a1.z, a1.w};
        float bv[8] = {b0.x, b0.y, b0.z, b0.w, b1.x, b1.y, b1.z, b1.w};
        #pragma unroll
        for (int ii = 0; ii < 8; ++ii) {
            #pragma unroll
            for (int jj = 0; jj < 8; ++jj) {
                // v_sub_f32 + v_add_f32 with |src| modifier: 2 VALU/term.
                acc[ii][jj] += __builtin_fabsf(av[ii] - bv[jj]);
            }
        }
    }

    // THETA = 1.0; |a/ls - b/ls| == |a-b| / |ls|  -> scale once at the end.
    const float scale = 1.0f / __builtin_fabsf(ls[0]);

    // ---- Epilogue: clamp, exp, non-temporal coalesced stores.
    #pragma unroll
    for (int a = 0; a < 8; ++a) {
        float* p = out + (size_t)(i0 + ti + a) * M + (j0 + tj);
        v4f r0, r1;
        r0.x = __expf(-fmaxf(acc[a][0] * scale, 1e-15f));
        r0.y = __expf(-fmaxf(acc[a][1] * scale, 1e-15f));
        r0.z = __expf(-fmaxf(acc[a][2] * scale, 1e-15f));
        r0.w = __expf(-fmaxf(acc[a][3] * scale, 1e-15f));
        r1.x = __expf(-fmaxf(acc[a][4] * scale, 1e-15f));
        r1.y = __expf(-fmaxf(acc[a][5] * scale, 1e-15f));
        r1.z = __expf(-fmaxf(acc[a][6] * scale, 1e-15f));
        r1.w = __expf(-fmaxf(acc[a][7] * scale, 1e-15f));
        // Write-once 256MB stream: NT hint keeps inputs resident in L2.
        __builtin_nontemporal_store(r0, (v4f*)p);
        __builtin_nontemporal_store(r1, (v4f*)(p + 4));
    }
}

extern "C" void kernel_launch(void* const* d_in, const int* in_sizes, int n_in,
                              void* d_out, int out_size, void* d_ws, size_t ws_size,
                              hipStream_t stream) {
    const float* x1 = (const float*)d_in[0];
    const float* x2 = (const float*)d_in[1];
    const float* ls = (const float*)d_in[2];
    float* out = (float*)d_out;

    const int N = in_sizes[0] / DIMS;   // 8192
    const int M = in_sizes[1] / DIMS;   // 8192

    dim3 grid(M / TILE, N / TILE);      // 64 x 64 = 4096 blocks
    dim3 block(256);                    // 8 wave32s
    laplace_l1_exp_kernel<<<grid, block, 0, stream>>>(x1, x2, ls, out, N, M);
}